// CARFACCell_3435973837112
// MI455X (gfx1250) — compile-verified
//
#include <hip/hip_runtime.h>

// Problem constants (from reference setup_inputs)
#define BB 2
#define CC 71
#define NN 1024
#define CP 80            // channel dim padded to 5 x 16 for WMMA tiles
#define MROWS (BB * NN)  // 2048 rows of the channel-mixing GEMM

typedef __attribute__((ext_vector_type(2))) float v2f;
typedef __attribute__((ext_vector_type(8))) float v8f;

// ---------------------------------------------------------------------------
// Kernel 1: build Wt[k][n] = (S^steps)[n][k] where S is the 71x71 one-step
// 3-tap symmetric-pad smoothing matrix. out[n] = sum_k in[k] * Wt[k][n].
// Rows k >= 71 of Wt are zero, so padded input channels never contribute.
// ---------------------------------------------------------------------------
__global__ void smooth_matrix_kernel(const float* __restrict__ kern,
                                     const int* __restrict__ steps_p,
                                     float* __restrict__ Wt) {
    __shared__ float bufA[CP * CP];
    __shared__ float bufB[CP * CP];
    const int t = threadIdx.x;
    const float k0 = kern[0], k1 = kern[1], k2 = kern[2];
    const int steps = steps_p[0];

    for (int i = t; i < CP * CP; i += blockDim.x) {
        int r = i / CP, c = i % CP;
        bufA[i] = (r == c && r < CC) ? 1.0f : 0.0f;
    }
    __syncthreads();

    float* src = bufA;
    float* dst = bufB;
    for (int s = 0; s < steps; ++s) {
        for (int i = t; i < CP * CP; i += blockDim.x) {
            int r = i / CP, c = i % CP;
            float v = 0.0f;
            if (r < CC && c < CC) {
                int rm = (r == 0) ? 0 : r - 1;          // symmetric reflect
                int rp = (r == CC - 1) ? CC - 1 : r + 1;
                v = k0 * src[rm * CP + c] + k1 * src[r * CP + c] +
                    k2 * src[rp * CP + c];
            }
            dst[i] = v;
        }
        __syncthreads();
        float* tmp = src; src = dst; dst = tmp;
    }
    // store transposed: Wt[k*CP + n] = M[n][k]  (robust to asymmetric taps)
    for (int i = t; i < CP * CP; i += blockDim.x) {
        int r = i / CP, c = i % CP;
        Wt[c * CP + r] = src[i];
    }
}

// ---------------------------------------------------------------------------
// Kernel 2: linear recurrence y[n] = f[n]*y[n-1] + g[n], one wave per (b,c)
// row. Each lane handles a 32-element chunk; chunks are affine maps
// (F = prod f, G = local response from 0) composed with a log2(32)-step
// wave-level inclusive shuffle scan. Output written pre-transposed as
// yt[(b*N + n)*CP + c] so the GEMM consumes it directly.
// ---------------------------------------------------------------------------
__global__ void recurrence_kernel(const float* __restrict__ a0,
                                  const float* __restrict__ f,
                                  const float* __restrict__ g,
                                  float* __restrict__ yt) {
    const int wave = (blockIdx.x * blockDim.x + threadIdx.x) >> 5;
    const int lane = threadIdx.x & 31;
    if (wave >= BB * CC) return;
    const int b = wave / CC;
    const int c = wave % CC;

    const float* fr = f + (size_t)wave * NN;
    const float* gr = g + (size_t)wave * NN;
    const int base = lane * 32;

    float ff[32], gg[32];
#pragma unroll
    for (int i = 0; i < 32; ++i) {
        ff[i] = fr[base + i];
        gg[i] = gr[base + i];
    }

    // Local affine map of this chunk: a_out = F * a_in + G
    float F = 1.0f, G = 0.0f;
#pragma unroll
    for (int i = 0; i < 32; ++i) {
        G = fmaf(ff[i], G, gg[i]);
        F *= ff[i];
    }

    // Inclusive scan of affine composition across lanes (earlier lanes first)
#pragma unroll
    for (int d = 1; d < 32; d <<= 1) {
        float Fp = __shfl_up(F, d, 32);
        float Gp = __shfl_up(G, d, 32);
        if (lane >= d) {
            G = fmaf(F, Gp, G);
            F = F * Fp;
        }
    }
    // Exclusive prefix -> state entering this chunk
    float Fe = __shfl_up(F, 1, 32);
    float Ge = __shfl_up(G, 1, 32);
    if (lane == 0) { Fe = 1.0f; Ge = 0.0f; }
    float a = fmaf(Fe, a0[wave], Ge);

    // Replay chunk, emitting outputs transposed to [B*N, CP]
#pragma unroll
    for (int i = 0; i < 32; ++i) {
        a = fmaf(ff[i], a, gg[i]);
        yt[((size_t)b * NN + (base + i)) * CP + c] = a;
    }
}

// ---------------------------------------------------------------------------
// Kernel 3: out_sm[m, n] = sum_k yt[m, k] * Wt[k, n] via V_WMMA_F32_16X16X4_F32.
// One wave per 16x16 output tile; 20 WMMA ops (K = 80 in steps of 4).
// Result scattered back to [B, C, N] layout.
// A frag (16x4 f32, 2 VGPR): lane -> M = lane%16, K = kb + 2*(lane/16) + {0,1}
// B frag (4x16 f32, 2 VGPR): lane -> N = lane%16, same K split
// C/D (8 VGPR): element r -> (M = r + 8*(lane/16), N = lane%16)
// ---------------------------------------------------------------------------
__global__ void gemm_smooth_kernel(const float* __restrict__ yt,
                                   const float* __restrict__ Wt,
                                   float* __restrict__ out) {
    const int waveId = (blockIdx.x * blockDim.x + threadIdx.x) >> 5;
    const int lane = threadIdx.x & 31;
    const int NT = CP / 16;                 // 5 tiles of channels
    const int MT = MROWS / 16;              // 128 tiles of (batch*time)
    if (waveId >= MT * NT) return;

    const int mt = waveId / NT;
    const int nt = waveId % NT;
    const int m0 = mt * 16;
    const int n0 = nt * 16;
    const int half = lane >> 4;             // 0 or 1
    const int ln = lane & 15;

    v8f acc = {0.f, 0.f, 0.f, 0.f, 0.f, 0.f, 0.f, 0.f};

#pragma unroll
    for (int kb = 0; kb < CP; kb += 4) {
        const int k0 = kb + 2 * half;
        // A: rows of yt (contiguous pair -> 64-bit load)
        v2f a = *(const v2f*)(yt + (size_t)(m0 + ln) * CP + k0);
        v2f b;
        b.x = Wt[(size_t)k0 * CP + n0 + ln];
        b.y = Wt[(size_t)(k0 + 1) * CP + n0 + ln];
        acc = __builtin_amdgcn_wmma_f32_16x16x4_f32(
            /*neg_a=*/false, a, /*neg_b=*/false, b,
            /*c_mod=*/(short)0, acc, /*reuse_a=*/false, /*reuse_b=*/false);
    }

    const int ch = n0 + ln;
#pragma unroll
    for (int r = 0; r < 8; ++r) {
        const int m = m0 + r + 8 * half;
        const int bb = m / NN;
        const int tt = m % NN;
        if (ch < CC)
            out[((size_t)bb * CC + ch) * NN + tt] = acc[r];
    }
}

// ---------------------------------------------------------------------------
extern "C" void kernel_launch(void* const* d_in, const int* in_sizes, int n_in,
                              void* d_out, int out_size, void* d_ws, size_t ws_size,
                              hipStream_t stream) {
    const float* a0    = (const float*)d_in[0];  // [B,C]
    const float* f     = (const float*)d_in[1];  // [B,C,N]
    const float* g     = (const float*)d_in[2];  // [B,C,N]
    const float* kern  = (const float*)d_in[3];  // [3]
    const int*   steps = (const int*)d_in[4];    // scalar

    float* out = (float*)d_out;                  // [B,C,N]

    // Workspace layout: Wt (80x80 f32, 25600 B, 256B-aligned) then yt (2048x80 f32)
    float* Wt = (float*)d_ws;
    float* yt = (float*)((char*)d_ws + (size_t)CP * CP * sizeof(float));

    // Zero yt so padded channel columns are clean on the very first call too.
    hipMemsetAsync(yt, 0, (size_t)MROWS * CP * sizeof(float), stream);

    smooth_matrix_kernel<<<1, 256, 0, stream>>>(kern, steps, Wt);

    // 142 waves needed; 36 blocks x 4 waves = 144 (extras exit as whole waves)
    recurrence_kernel<<<36, 128, 0, stream>>>(a0, f, g, yt);

    // 640 tiles = 640 waves; 160 blocks x 4 waves, exact fit (EXEC all-1s at WMMA)
    gemm_smooth_kernel<<<160, 128, 0, stream>>>(yt, Wt, out);
}